// PMLP_GCNII_79353815761146
// MI455X (gfx1250) — compile-verified
//
#include <hip/hip_runtime.h>
#include <hip/hip_bf16.h>

typedef __attribute__((ext_vector_type(2))) float v2f;
typedef __attribute__((ext_vector_type(8))) float v8f;

#define EPSBN 1e-5f

// ---------------- elementwise / graph-normalization kernels ----------------

__global__ void zero_f(float* __restrict__ p, long long n) {
  long long i = (long long)blockIdx.x * blockDim.x + threadIdx.x;
  if (i < n) p[i] = 0.0f;
}

__global__ void deg_edges(const long long* __restrict__ src,
                          const long long* __restrict__ dst,
                          float* __restrict__ deg, int E) {
  int e = blockIdx.x * blockDim.x + threadIdx.x;
  if (e < E) {
    int s = (int)src[e], d = (int)dst[e];
    if (s != d) atomicAdd(&deg[d], 1.0f);
  }
}

// deg_dinv holds raw non-self-loop degree on entry; on exit holds deg^-1/2.
__global__ void finalize_deg(float* __restrict__ deg_dinv,
                             float* __restrict__ self_w, int n) {
  int i = blockIdx.x * blockDim.x + threadIdx.x;
  if (i < n) {
    float d = deg_dinv[i] + 1.0f;   // +1 for added self-loop
    deg_dinv[i] = rsqrtf(d);
    self_w[i] = 1.0f / d;
  }
}

__global__ void edge_w(const long long* __restrict__ src,
                       const long long* __restrict__ dst,
                       const float* __restrict__ dinv,
                       float* __restrict__ w, int E) {
  int e = blockIdx.x * blockDim.x + threadIdx.x;
  if (e < E) {
    int s = (int)src[e], d = (int)dst[e];
    w[e] = (s != d) ? dinv[s] * dinv[d] : 0.0f;
  }
}

// ---------------- WMMA GEMM: Out = aCoef*Hin + bCoef*(A @ W^T) ----------------
// A: [M x K] row-major. W: [CN x K] row-major. Out/Hin: [M x CN] row-major.
// Per ISA 7.12.2 (f32 16x16x4): A lane=M(0..15), half=K-pair; B lane=N, half=K-pair;
// D vgpr v -> M = v + half*8, N = lane&15.
//
// Out-of-range rows are clamped to row 0 for the loads; their (garbage/duplicate)
// results stay confined to D rows we never write (m < M guard), so no zero-masking
// is needed -> A traffic is one unconditional global_load_b64 per k-step.
template <int K, int CN, int CPAD>
__global__ __launch_bounds__(256) void gemm_wmma_kern(
    const float* __restrict__ A, const float* __restrict__ W,
    const float* __restrict__ Hin, float* __restrict__ Out,
    int M, float aCoef, float bCoef) {
  constexpr int NT = CPAD / 16;
  __shared__ float sW[CN * K];          // <= 48 KB (fits easily in 320 KB WGP LDS)
  for (int i = threadIdx.x; i < CN * K; i += 256) sW[i] = W[i];
  __syncthreads();

  const int wave = threadIdx.x >> 5;    // 8 wave32 per block
  const int lane = threadIdx.x & 31;
  const int half = lane >> 4;
  const int l15 = lane & 15;
  const int m0 = blockIdx.x * 128 + wave * 16;

  const int arow = m0 + l15;
  const float* Arow = A + (size_t)(arow < M ? arow : 0) * K;

  v8f acc[NT] = {};

  for (int k = 0; k < K; k += 4) {
    const int ka = k + half * 2;
    const v2f a = *(const v2f*)(Arow + ka);   // single b64 load, no predication
#pragma unroll
    for (int t = 0; t < NT; ++t) {
      const int n = t * 16 + l15;
      v2f b;
      if (n < CN) {
        b.x = sW[n * K + ka];
        b.y = sW[n * K + ka + 1];
      } else {
        b.x = 0.0f;
        b.y = 0.0f;
      }
      acc[t] = __builtin_amdgcn_wmma_f32_16x16x4_f32(
          false, a, false, b, (short)0, acc[t], false, false);
    }
  }

#pragma unroll
  for (int t = 0; t < NT; ++t) {
    const int n = t * 16 + l15;
    if (n >= CN) continue;
#pragma unroll
    for (int v = 0; v < 8; ++v) {
      const int m = m0 + half * 8 + v;
      if (m < M) {
        const size_t o = (size_t)m * CN + n;
        float r = bCoef * acc[t][v];
        if (aCoef != 0.0f) r += aCoef * Hin[o];
        Out[o] = r;
      }
    }
  }
}

// ---------------- SpMM: out = self_w[row]*t  then  out[dst] += w_e * t[src] ----------------

__global__ void spmm_init(const float* __restrict__ t, const float* __restrict__ selfw,
                          float* __restrict__ out, long long total, int C) {
  long long idx = (long long)blockIdx.x * blockDim.x + threadIdx.x;
  if (idx < total) out[idx] = selfw[idx / C] * t[idx];
}

__global__ void spmm_edge(const long long* __restrict__ src,
                          const long long* __restrict__ dst,
                          const float* __restrict__ w,
                          const float* __restrict__ h,
                          float* __restrict__ out, int E, int C) {
  int e = blockIdx.x * blockDim.x + threadIdx.x;
  if (e >= E) return;
  float we = w[e];
  if (we == 0.0f) return;           // self-loop edges contribute nothing
  long long s = src[e], d = dst[e];
  const float4* hs = (const float4*)(h + s * (long long)C);
  float* od = out + d * (long long)C;
  const int c4 = C >> 2;
  for (int c = 0; c < c4; ++c) {
    float4 v = hs[c];
    atomicAdd(&od[4 * c + 0], we * v.x);
    atomicAdd(&od[4 * c + 1], we * v.y);
    atomicAdd(&od[4 * c + 2], we * v.z);
    atomicAdd(&od[4 * c + 3], we * v.w);
  }
}

// ---------------- residual blend + BN stats (one pass), then BN+ReLU ----------------

__global__ void combine_stats(const float* __restrict__ s, const float* __restrict__ h0,
                              float* __restrict__ u, float ca, float cb,
                              long long total, int C,
                              float* __restrict__ sum, float* __restrict__ sumsq) {
  __shared__ float ls[96], ls2[96];
  for (int i = threadIdx.x; i < C; i += blockDim.x) { ls[i] = 0.0f; ls2[i] = 0.0f; }
  __syncthreads();
  const long long stride = (long long)gridDim.x * blockDim.x;
  for (long long idx = (long long)blockIdx.x * blockDim.x + threadIdx.x; idx < total;
       idx += stride) {
    float v = ca * s[idx];
    if (cb != 0.0f) v += cb * h0[idx];
    u[idx] = v;
    int c = (int)(idx % C);
    atomicAdd(&ls[c], v);        // ds_add_f32, low-conflict (consecutive lanes -> distinct c)
    atomicAdd(&ls2[c], v * v);
  }
  __syncthreads();
  for (int i = threadIdx.x; i < C; i += blockDim.x) {
    atomicAdd(&sum[i], ls[i]);
    atomicAdd(&sumsq[i], ls2[i]);
  }
}

__global__ void bn_relu(const float* __restrict__ u, float* __restrict__ out,
                        const float* __restrict__ sum, const float* __restrict__ sumsq,
                        long long total, int C, float invn) {
  long long idx = (long long)blockIdx.x * blockDim.x + threadIdx.x;
  if (idx < total) {
    int c = (int)(idx % C);
    float mu = sum[c] * invn;
    float var = sumsq[c] * invn - mu * mu;   // biased variance, as torch BN
    float v = (u[idx] - mu) * rsqrtf(var + EPSBN);
    out[idx] = v > 0.0f ? v : 0.0f;
  }
}

// ---------------- driver ----------------

extern "C" void kernel_launch(void* const* d_in, const int* in_sizes, int n_in,
                              void* d_out, int out_size, void* d_ws, size_t ws_size,
                              hipStream_t stream) {
  const float* x = (const float*)d_in[0];
  const long long* ei = (const long long*)d_in[1];   // int64 edge_index [2, E]
  const float* W0 = (const float*)d_in[2];
  const float* W1 = (const float*)d_in[3];
  const float* W2 = (const float*)d_in[4];
  const float* W3 = (const float*)d_in[5];
  float* out = (float*)d_out;

  const int N = in_sizes[0] / 128;
  const int E = in_sizes[1] / 2;
  const long long* srcv = ei;
  const long long* dstv = ei + E;

  char* base = (char*)d_ws;
  size_t off = 0;
  auto alloc = [&](size_t bytes) -> float* {
    size_t s = (off + 255) & ~(size_t)255;
    float* p = (float*)(base + s);
    off = s + bytes;
    return p;
  };
  float* dinv = alloc((size_t)N * 4);          // deg accumulator, then deg^-1/2
  float* selfw = alloc((size_t)N * 4);
  float* ew = alloc((size_t)E * 4);
  float* bufA = alloc((size_t)N * 96 * 4);     // h0 lives here
  float* bufB = alloc((size_t)N * 96 * 4);
  float* bufC = alloc((size_t)N * 96 * 4);
  float* buf40 = alloc((size_t)N * 40 * 4);
  float* stat = alloc(2 * 96 * 4);
  float* ssum = stat;
  float* ssq = stat + 96;
  (void)ws_size; (void)n_in; (void)out_size;

  const int T = 256;
  const int gN = (N + T - 1) / T;
  const int gE = (E + T - 1) / T;
  const long long nh = (long long)N * 96;
  const int gNH = (int)((nh + T - 1) / T);
  const long long n40 = (long long)N * 40;
  const int gN40 = (int)((n40 + T - 1) / T);
  const int gG = (N + 127) / 128;
  const float invn = 1.0f / (float)N;

  // --- symmetric normalization weights ---
  zero_f<<<gN, T, 0, stream>>>(dinv, N);
  deg_edges<<<gE, T, 0, stream>>>(srcv, dstv, dinv, E);
  finalize_deg<<<gN, T, 0, stream>>>(dinv, selfw, N);
  edge_w<<<gE, T, 0, stream>>>(srcv, dstv, dinv, ew, E);

  // --- layer 0: h0 = relu(bn(x @ W0^T)) ---
  gemm_wmma_kern<128, 96, 96><<<gG, T, 0, stream>>>(x, W0, x, bufC, N, 0.0f, 1.0f);
  zero_f<<<1, T, 0, stream>>>(stat, 192);
  combine_stats<<<1024, T, 0, stream>>>(bufC, bufC, bufC, 1.0f, 0.0f, nh, 96, ssum, ssq);
  bn_relu<<<gNH, T, 0, stream>>>(bufC, bufA, ssum, ssq, nh, 96, invn);  // h = h0 = bufA

  // --- layer 1 (beta = 0.5): t = 0.5*h + 0.5*h@W1^T ; u = 0.9*spmm(t) + 0.1*h0 ---
  gemm_wmma_kern<96, 96, 96><<<gG, T, 0, stream>>>(bufA, W1, bufA, bufC, N, 0.5f, 0.5f);
  spmm_init<<<gNH, T, 0, stream>>>(bufC, selfw, bufB, nh, 96);
  spmm_edge<<<gE, T, 0, stream>>>(srcv, dstv, ew, bufC, bufB, E, 96);
  zero_f<<<1, T, 0, stream>>>(stat, 192);
  combine_stats<<<1024, T, 0, stream>>>(bufB, bufA, bufB, 0.9f, 0.1f, nh, 96, ssum, ssq);
  bn_relu<<<gNH, T, 0, stream>>>(bufB, bufC, ssum, ssq, nh, 96, invn);  // h = bufC

  // --- layer 2 (beta = 0.25) ---
  gemm_wmma_kern<96, 96, 96><<<gG, T, 0, stream>>>(bufC, W2, bufC, bufB, N, 0.75f, 0.25f);
  spmm_init<<<gNH, T, 0, stream>>>(bufB, selfw, bufC, nh, 96);
  spmm_edge<<<gE, T, 0, stream>>>(srcv, dstv, ew, bufB, bufC, E, 96);
  zero_f<<<1, T, 0, stream>>>(stat, 192);
  combine_stats<<<1024, T, 0, stream>>>(bufC, bufA, bufC, 0.9f, 0.1f, nh, 96, ssum, ssq);
  bn_relu<<<gNH, T, 0, stream>>>(bufC, bufB, ssum, ssq, nh, 96, invn);  // h = bufB

  // --- output layer: out = spmm(h @ W3^T), accumulated directly in d_out ---
  gemm_wmma_kern<96, 40, 48><<<gG, T, 0, stream>>>(bufB, W3, bufB, buf40, N, 0.0f, 1.0f);
  spmm_init<<<gN40, T, 0, stream>>>(buf40, selfw, out, n40, 40);
  spmm_edge<<<gE, T, 0, stream>>>(srcv, dstv, ew, buf40, out, E, 40);
}